// Hessian5_16501264351429
// MI455X (gfx1250) — compile-verified
//
#include <hip/hip_runtime.h>
#include <math.h>

// ---------------------------------------------------------------------------
// MI455X / gfx1250, wave32. f32 data -> V_WMMA_F32_16X16X4_F32 throughout.
// A-frag (16x4): lane L holds A[m=L&15][kb], A[m][kb+1], kb = k0 + 2*(L>>4)
// B-frag (4x16): lane L holds B[kb][n=L&15], B[kb+1][n]
// C/D (16x16):   VGPR r, lane L -> C[m = r + 8*(L>>4)][n = L&15]
// Pairwise weights staged to LDS once per WG via async global->LDS copies
// (ASYNCcnt path), per CDNA5 ISA.
// ---------------------------------------------------------------------------

typedef float v2f __attribute__((ext_vector_type(2)));
typedef float v8f __attribute__((ext_vector_type(8)));

typedef __attribute__((address_space(1))) int as1_int;
typedef __attribute__((address_space(3))) int as3_int;

__device__ __forceinline__ v8f wmma4(v2f a, v2f b, v8f c) {
  return __builtin_amdgcn_wmma_f32_16x16x4_f32(false, a, false, b, (short)0, c,
                                               false, false);
}

__device__ __forceinline__ float silu_f(float x) {
  return x / (1.0f + __expf(-x));
}

#define ZERO8 (v8f){0.f,0.f,0.f,0.f,0.f,0.f,0.f,0.f}

// ---- async global -> LDS copy (GLOBAL_LOAD_ASYNC_TO_LDS_B32) --------------
__device__ __forceinline__ void async_copy_f32(float* lds_dst, const float* gsrc) {
#if __has_builtin(__builtin_amdgcn_global_load_async_to_lds_b32)
  __builtin_amdgcn_global_load_async_to_lds_b32(
      (as1_int*)gsrc, (as3_int*)lds_dst, 0, 0);
#else
  *lds_dst = *gsrc;   // synchronous fallback
#endif
}

__device__ __forceinline__ void wait_async0() {
#if __has_builtin(__builtin_amdgcn_s_wait_asynccnt)
  __builtin_amdgcn_s_wait_asynccnt(0);
#else
  asm volatile("s_wait_asynccnt 0x0" ::: "memory");
#endif
}

// ---------------------------------------------------------------------------
// Generic row-major GEMM: C[M,N] = act(A[M,K] @ W[K,N] + bias)
// blockDim = 256 (8 waves); wave w owns n-tile w (requires N <= 128, N%16==0).
// grid.x = M/16. K % 4 == 0.
// ---------------------------------------------------------------------------
__global__ __launch_bounds__(256) void gemm_wmma_f32(
    const float* __restrict__ A, const float* __restrict__ W,
    const float* __restrict__ bias, float* __restrict__ C,
    int N, int K, int apply_silu)
{
  const int lane = threadIdx.x & 31;
  const int wave = threadIdx.x >> 5;
  const int half = lane >> 4;
  const int l15  = lane & 15;
  if (wave * 16 >= N) return;            // wave-uniform exit, EXEC stays full

  const int row = blockIdx.x * 16 + l15; // A row this lane feeds
  const int col = wave * 16 + l15;       // C column this lane owns

  v8f acc = ZERO8;
  for (int k0 = 0; k0 < K; k0 += 4) {
    const int ka = k0 + 2 * half;
    v2f a; a.x = A[(size_t)row * K + ka];     a.y = A[(size_t)row * K + ka + 1];
    v2f b; b.x = W[(size_t)ka  * N + col];    b.y = W[(size_t)(ka + 1) * N + col];
    acc = wmma4(a, b, acc);
  }
  const float bv = bias ? bias[col] : 0.0f;
#pragma unroll
  for (int r = 0; r < 8; ++r) {
    const int m = blockIdx.x * 16 + r + 8 * half;
    float v = acc[r] + bv;
    if (apply_silu) v = silu_f(v);
    C[(size_t)m * N + col] = v;
  }
}

// ---------------------------------------------------------------------------
// xin[n, 0:128] = s[n];  xin[n, 128+o] = || vmix[n, :, o] ||  (o < 64)
// ---------------------------------------------------------------------------
__global__ void k_vvn_concat(const float* __restrict__ s,
                             const float* __restrict__ vmix,
                             float* __restrict__ xin, int N)
{
  int t = blockIdx.x * blockDim.x + threadIdx.x;
  if (t >= N * 192) return;
  int n = t / 192, j = t - n * 192;
  float v;
  if (j < 128) {
    v = s[(size_t)n * 128 + j];
  } else {
    int o = j - 128;
    float a = vmix[((size_t)n * 3 + 0) * 128 + o];
    float b = vmix[((size_t)n * 3 + 1) * 128 + o];
    float c = vmix[((size_t)n * 3 + 2) * 128 + o];
    v = sqrtf(a * a + b * b + c * c);
  }
  xin[t] = v;
}

// ---------------------------------------------------------------------------
// s1 = silu(x2[:, :64]); v1[n,d,o] = x2[n, 64+o] * vmix[n, d, 64+o]
// ---------------------------------------------------------------------------
__global__ void k_epilogue1(const float* __restrict__ x2,
                            const float* __restrict__ vmix,
                            float* __restrict__ s1, float* __restrict__ v1,
                            int N)
{
  int t = blockIdx.x * blockDim.x + threadIdx.x;
  if (t >= N * 64) return;
  int n = t >> 6, o = t & 63;
  s1[t] = silu_f(x2[(size_t)n * 128 + o]);
  float g = x2[(size_t)n * 128 + 64 + o];
#pragma unroll
  for (int d = 0; d < 3; ++d)
    v1[((size_t)n * 3 + d) * 64 + o] = g * vmix[((size_t)n * 3 + d) * 128 + 64 + o];
}

// ---------------------------------------------------------------------------
// Stage-2 gated block, one thread per atom (tiny: 65->64->2).
// ---------------------------------------------------------------------------
__global__ void k_stage2(const float* __restrict__ s1, const float* __restrict__ v1,
                         const float* __restrict__ mixW2,
                         const float* __restrict__ W1, const float* __restrict__ b1,
                         const float* __restrict__ W2, const float* __restrict__ b2,
                         float* __restrict__ l0, float* __restrict__ l1, int N)
{
  int n = blockIdx.x * blockDim.x + threadIdx.x;
  if (n >= N) return;

  float vm[3][2] = {{0, 0}, {0, 0}, {0, 0}};
  for (int o = 0; o < 64; ++o) {
    float w0 = mixW2[o * 2 + 0], w1 = mixW2[o * 2 + 1];
#pragma unroll
    for (int d = 0; d < 3; ++d) {
      float val = v1[((size_t)n * 3 + d) * 64 + o];
      vm[d][0] += val * w0;
      vm[d][1] += val * w1;
    }
  }
  float vn = sqrtf(vm[0][0] * vm[0][0] + vm[1][0] * vm[1][0] + vm[2][0] * vm[2][0]);

  float h[64];
  for (int j = 0; j < 64; ++j) h[j] = b1[j] + vn * W1[64 * 64 + j];
  for (int k = 0; k < 64; ++k) {
    float sv = s1[(size_t)n * 64 + k];
    for (int j = 0; j < 64; ++j) h[j] += sv * W1[k * 64 + j];
  }
  float o0 = b2[0], o1 = b2[1];
  for (int j = 0; j < 64; ++j) {
    float hs = silu_f(h[j]);
    o0 += hs * W2[j * 2 + 0];
    o1 += hs * W2[j * 2 + 1];
  }
  l0[n] = o0;
#pragma unroll
  for (int d = 0; d < 3; ++d) l1[(size_t)n * 3 + d] = o1 * vm[d][1];
}

// ---------------------------------------------------------------------------
// Pairwise stage. Each wave owns a 16-pair tile: fixed (b, i), j = j0..j0+15.
// Per tile: vv/vr/sp MLPs (9|2 -> 30 -> 9) summed, then h MLP (9 -> 30 -> 9).
// All MLP weights staged to LDS once per workgroup via async copies.
// ---------------------------------------------------------------------------
#define LH 36   // hidden LDS row stride (30 padded to 32, +conflict pad)
#define LT 13   // t-matrix LDS row stride (9 padded to 12, +conflict pad)

// LDS weight-pack offsets (floats)
#define O_VVW1 0
#define O_VVB1 270
#define O_VVW2 300
#define O_VVB2 570
#define O_VRW1 579
#define O_VRB1 849
#define O_VRW2 879
#define O_VRB2 1149
#define O_SW1  1158
#define O_SB1  1218
#define O_SW2  1248
#define O_SB2  1518
#define O_HW1  1527
#define O_HB1  1797
#define O_HW2  1827
#define O_HB2  2097
#define W_TOTAL 2106

__global__ __launch_bounds__(256) void k_pairwise(
    const float* __restrict__ l0g, const float* __restrict__ l1g,
    const float* __restrict__ pos,
    const float* __restrict__ vvW1, const float* __restrict__ vvB1,
    const float* __restrict__ vvW2, const float* __restrict__ vvB2,
    const float* __restrict__ vrW1, const float* __restrict__ vrB1,
    const float* __restrict__ vrW2, const float* __restrict__ vrB2,
    const float* __restrict__ sW1,  const float* __restrict__ sB1,
    const float* __restrict__ sW2,  const float* __restrict__ sB2,
    const float* __restrict__ hW1,  const float* __restrict__ hB1,
    const float* __restrict__ hW2,  const float* __restrict__ hB2,
    float* __restrict__ out)
{
  __shared__ float wb[W_TOTAL];     // staged weight pack
  __shared__ float ldsH[8][16 * LH];
  __shared__ float ldsT[8][16 * LT];

  const int lane = threadIdx.x & 31;
  const int wave = threadIdx.x >> 5;
  const int half = lane >> 4;
  const int l15  = lane & 15;
  const int b    = blockIdx.x / 6;
  const int j0   = (blockIdx.x % 6) * 16;

  float* myH = ldsH[wave];
  float* myT = ldsT[wave];

  // ---- stage all weights/biases into LDS with async global->LDS copies ----
  {
    const float* srcs[16] = { vvW1, vvB1, vvW2, vvB2,  vrW1, vrB1, vrW2, vrB2,
                              sW1,  sB1,  sW2,  sB2,   hW1,  hB1,  hW2,  hB2 };
    const int offs[16] = { O_VVW1, O_VVB1, O_VVW2, O_VVB2,
                           O_VRW1, O_VRB1, O_VRW2, O_VRB2,
                           O_SW1,  O_SB1,  O_SW2,  O_SB2,
                           O_HW1,  O_HB1,  O_HW2,  O_HB2 };
    const int lens[16] = { 270, 30, 270, 9,  270, 30, 270, 9,
                           60,  30, 270, 9,  270, 30, 270, 9 };
    for (int a = 0; a < 16; ++a)
      for (int idx = threadIdx.x; idx < lens[a]; idx += 256)
        async_copy_f32(&wb[offs[a] + idx], &srcs[a][idx]);
    wait_async0();
    __syncthreads();
  }

  // ---- layer-1 weight fragments: K in {9,2} padded to 12/4, N=30 pad 32 ----
  v2f fvv1[3][2], fvr1[3][2], fh1[3][2], fs1v[2];
#pragma unroll
  for (int t = 0; t < 3; ++t) {
#pragma unroll
    for (int nt = 0; nt < 2; ++nt) {
      int kb = 4 * t + 2 * half;
      int c  = nt * 16 + l15;
      v2f w;
      w.x = (kb     < 9 && c < 30) ? wb[O_VVW1 + kb * 30 + c]       : 0.f;
      w.y = (kb + 1 < 9 && c < 30) ? wb[O_VVW1 + (kb + 1) * 30 + c] : 0.f;
      fvv1[t][nt] = w;
      w.x = (kb     < 9 && c < 30) ? wb[O_VRW1 + kb * 30 + c]       : 0.f;
      w.y = (kb + 1 < 9 && c < 30) ? wb[O_VRW1 + (kb + 1) * 30 + c] : 0.f;
      fvr1[t][nt] = w;
      w.x = (kb     < 9 && c < 30) ? wb[O_HW1 + kb * 30 + c]        : 0.f;
      w.y = (kb + 1 < 9 && c < 30) ? wb[O_HW1 + (kb + 1) * 30 + c]  : 0.f;
      fh1[t][nt] = w;
    }
  }
#pragma unroll
  for (int nt = 0; nt < 2; ++nt) {
    int kb = 2 * half;
    int c  = nt * 16 + l15;
    v2f w;
    w.x = (kb     < 2 && c < 30) ? wb[O_SW1 + kb * 30 + c]       : 0.f;
    w.y = (kb + 1 < 2 && c < 30) ? wb[O_SW1 + (kb + 1) * 30 + c] : 0.f;
    fs1v[nt] = w;
  }
  // layer-1 biases (per hidden column of this lane)
  float bvv1[2], bvr1[2], bs1b[2], bh1b[2];
#pragma unroll
  for (int nt = 0; nt < 2; ++nt) {
    int c = nt * 16 + l15;
    bvv1[nt] = (c < 30) ? wb[O_VVB1 + c] : 0.f;
    bvr1[nt] = (c < 30) ? wb[O_VRB1 + c] : 0.f;
    bs1b[nt] = (c < 30) ? wb[O_SB1  + c] : 0.f;
    bh1b[nt] = (c < 30) ? wb[O_HB1  + c] : 0.f;
  }
  // ---- layer-2 weight fragments: K=30 pad 32, N=9 pad 16 ----
  v2f fvv2[8], fvr2[8], fs2[8], fh2[8];
#pragma unroll
  for (int t = 0; t < 8; ++t) {
    int kb = 4 * t + 2 * half;
    int c  = l15;
    v2f w;
    w.x = (kb     < 30 && c < 9) ? wb[O_VVW2 + kb * 9 + c]       : 0.f;
    w.y = (kb + 1 < 30 && c < 9) ? wb[O_VVW2 + (kb + 1) * 9 + c] : 0.f;
    fvv2[t] = w;
    w.x = (kb     < 30 && c < 9) ? wb[O_VRW2 + kb * 9 + c]       : 0.f;
    w.y = (kb + 1 < 30 && c < 9) ? wb[O_VRW2 + (kb + 1) * 9 + c] : 0.f;
    fvr2[t] = w;
    w.x = (kb     < 30 && c < 9) ? wb[O_SW2 + kb * 9 + c]        : 0.f;
    w.y = (kb + 1 < 30 && c < 9) ? wb[O_SW2 + (kb + 1) * 9 + c]  : 0.f;
    fs2[t] = w;
    w.x = (kb     < 30 && c < 9) ? wb[O_HW2 + kb * 9 + c]        : 0.f;
    w.y = (kb + 1 < 30 && c < 9) ? wb[O_HW2 + (kb + 1) * 9 + c]  : 0.f;
    fh2[t] = w;
  }
  const float tb2 = (l15 < 9)
      ? (wb[O_VVB2 + l15] + wb[O_VRB2 + l15] + wb[O_SB2 + l15]) : 0.f;
  const float hb2 = (l15 < 9) ? wb[O_HB2 + l15] : 0.f;

  // ---- per-lane j data (pair m <-> j = j0 + m, m = l15) ----
  const int jv = j0 + l15;
  const size_t nj = (size_t)b * 96 + jv;
  const float l1j[3] = { l1g[nj * 3 + 0], l1g[nj * 3 + 1], l1g[nj * 3 + 2] };
  const float pj [3] = { pos[nj * 3 + 0], pos[nj * 3 + 1], pos[nj * 3 + 2] };
  const float l0j = l0g[nj];

  for (int i = wave; i < 96; i += 8) {
    const size_t ni = (size_t)b * 96 + i;
    const float l1i[3] = { l1g[ni * 3 + 0], l1g[ni * 3 + 1], l1g[ni * 3 + 2] };
    const float l0i = l0g[ni];

    // layer-1 A rows in registers (K padded to 12)
    float rowvv[12], rowvr[12];
#pragma unroll
    for (int k = 0; k < 12; ++k) {
      rowvv[k] = (k < 9) ? l1i[k / 3] * l1j[k % 3] : 0.f;
      rowvr[k] = (k < 9) ? l1i[k / 3] * pj [k % 3] : 0.f;
    }

    v8f acc2 = ZERO8;  // sum of vv/vr/sp layer-2 outputs (the "t" matrix)

#pragma unroll
    for (int mlp = 0; mlp < 3; ++mlp) {
      // ---- layer 1 ----
      v8f c0 = ZERO8, c1 = ZERO8;
      if (mlp < 2) {
        const float* row = (mlp == 0) ? rowvv : rowvr;
        const v2f (*fw)[2] = (mlp == 0) ? fvv1 : fvr1;
#pragma unroll
        for (int t = 0; t < 3; ++t) {
          v2f a;
          a.x = half ? row[4 * t + 2] : row[4 * t + 0];
          a.y = half ? row[4 * t + 3] : row[4 * t + 1];
          c0 = wmma4(a, fw[t][0], c0);
          c1 = wmma4(a, fw[t][1], c1);
        }
      } else {  // sp MLP: K = 2 (pad 4) -> single k-step
        v2f a;
        a.x = half ? 0.f : l0i;
        a.y = half ? 0.f : l0j;
        c0 = wmma4(a, fs1v[0], c0);
        c1 = wmma4(a, fs1v[1], c1);
      }
      // ---- silu + bias -> LDS (16 x 32, stride LH), zero-pad cols 30..31 ----
      const float bb0 = (mlp == 0) ? bvv1[0] : (mlp == 1) ? bvr1[0] : bs1b[0];
      const float bb1 = (mlp == 0) ? bvv1[1] : (mlp == 1) ? bvr1[1] : bs1b[1];
#pragma unroll
      for (int r = 0; r < 8; ++r) {
        int m = r + 8 * half;
        myH[m * LH + l15] = silu_f(c0[r] + bb0);
        int c16 = 16 + l15;
        myH[m * LH + c16] = (c16 < 30) ? silu_f(c1[r] + bb1) : 0.f;
      }
      // ---- layer 2: K=30 pad 32, N=9 pad 16, accumulate into acc2 ----
      const v2f* fw2 = (mlp == 0) ? fvv2 : (mlp == 1) ? fvr2 : fs2;
#pragma unroll
      for (int t = 0; t < 8; ++t) {
        int ka = 4 * t + 2 * half;
        v2f a;
        a.x = myH[l15 * LH + ka];
        a.y = myH[l15 * LH + ka + 1];
        acc2 = wmma4(a, fw2[t], acc2);
      }
    }

    // ---- t matrix (+bias) -> LDS transpose buffer (16 x 12, stride LT) ----
#pragma unroll
    for (int r = 0; r < 8; ++r) {
      int m = r + 8 * half;
      if (l15 < 9)       myT[m * LT + l15] = acc2[r] + tb2;
      else if (l15 < 12) myT[m * LT + l15] = 0.f;
    }

    // ---- h MLP layer 1 ----
    v8f c0 = ZERO8, c1 = ZERO8;
#pragma unroll
    for (int t = 0; t < 3; ++t) {
      int ka = 4 * t + 2 * half;
      v2f a;
      a.x = myT[l15 * LT + ka];
      a.y = myT[l15 * LT + ka + 1];
      c0 = wmma4(a, fh1[t][0], c0);
      c1 = wmma4(a, fh1[t][1], c1);
    }
#pragma unroll
    for (int r = 0; r < 8; ++r) {
      int m = r + 8 * half;
      myH[m * LH + l15] = silu_f(c0[r] + bh1b[0]);
      int c16 = 16 + l15;
      myH[m * LH + c16] = (c16 < 30) ? silu_f(c1[r] + bh1b[1]) : 0.f;
    }
    // ---- h MLP layer 2 ----
    v8f ho = ZERO8;
#pragma unroll
    for (int t = 0; t < 8; ++t) {
      int ka = 4 * t + 2 * half;
      v2f a;
      a.x = myH[l15 * LH + ka];
      a.y = myH[l15 * LH + ka + 1];
      ho = wmma4(a, fh2[t], ho);
    }
    // ---- store: out[b][(i*3+k)][(j0+m)*3+l], col = k*3+l ----
    if (l15 < 9) {
      const int k = l15 / 3, l = l15 % 3;
#pragma unroll
      for (int r = 0; r < 8; ++r) {
        int m = r + 8 * half;
        out[((size_t)b * 288 + (i * 3 + k)) * 288 + (size_t)(j0 + m) * 3 + l] =
            ho[r] + hb2;
      }
    }
  }
}

// ---------------------------------------------------------------------------
extern "C" void kernel_launch(void* const* d_in, const int* in_sizes, int n_in,
                              void* d_out, int out_size, void* d_ws, size_t ws_size,
                              hipStream_t stream)
{
  (void)in_sizes; (void)n_in; (void)out_size; (void)ws_size;
  const int N = 12288;           // B*A
  const float* positions = (const float*)d_in[0];
  const float* scalar    = (const float*)d_in[1];
  const float* vector    = (const float*)d_in[2];
  const float* mix_W1    = (const float*)d_in[3];
  const float* sc1_W1    = (const float*)d_in[4];
  const float* sc1_b1    = (const float*)d_in[5];
  const float* sc1_W2    = (const float*)d_in[6];
  const float* sc1_b2    = (const float*)d_in[7];
  const float* mix_W2    = (const float*)d_in[8];
  const float* sc2_W1    = (const float*)d_in[9];
  const float* sc2_b1    = (const float*)d_in[10];
  const float* sc2_W2    = (const float*)d_in[11];
  const float* sc2_b2    = (const float*)d_in[12];
  const float* vv_W1 = (const float*)d_in[13]; const float* vv_b1 = (const float*)d_in[14];
  const float* vv_W2 = (const float*)d_in[15]; const float* vv_b2 = (const float*)d_in[16];
  const float* vr_W1 = (const float*)d_in[17]; const float* vr_b1 = (const float*)d_in[18];
  const float* vr_W2 = (const float*)d_in[19]; const float* vr_b2 = (const float*)d_in[20];
  const float* s_W1  = (const float*)d_in[21]; const float* s_b1  = (const float*)d_in[22];
  const float* s_W2  = (const float*)d_in[23]; const float* s_b2  = (const float*)d_in[24];
  const float* h_W1  = (const float*)d_in[25]; const float* h_b1  = (const float*)d_in[26];
  const float* h_W2  = (const float*)d_in[27]; const float* h_b2  = (const float*)d_in[28];
  float* out = (float*)d_out;

  float* ws = (float*)d_ws;
  size_t off = 0;
  float* vmix = ws + off; off += (size_t)N * 3 * 128;  // 36864 x 128
  float* xin  = ws + off; off += (size_t)N * 192;
  float* h1   = ws + off; off += (size_t)N * 128;
  float* x2   = ws + off; off += (size_t)N * 128;
  float* s1   = ws + off; off += (size_t)N * 64;
  float* v1   = ws + off; off += (size_t)N * 3 * 64;
  float* l0   = ws + off; off += (size_t)N;
  float* l1v  = ws + off; off += (size_t)N * 3;

  // Stage 1
  gemm_wmma_f32<<<(N * 3) / 16, 256, 0, stream>>>(vector, mix_W1, nullptr, vmix,
                                                  128, 128, 0);
  k_vvn_concat<<<(N * 192 + 255) / 256, 256, 0, stream>>>(scalar, vmix, xin, N);
  gemm_wmma_f32<<<N / 16, 256, 0, stream>>>(xin, sc1_W1, sc1_b1, h1, 128, 192, 1);
  gemm_wmma_f32<<<N / 16, 256, 0, stream>>>(h1, sc1_W2, sc1_b2, x2, 128, 128, 0);
  k_epilogue1<<<(N * 64 + 255) / 256, 256, 0, stream>>>(x2, vmix, s1, v1, N);

  // Stage 2 (tiny)
  k_stage2<<<(N + 127) / 128, 128, 0, stream>>>(s1, v1, mix_W2, sc2_W1, sc2_b1,
                                                sc2_W2, sc2_b2, l0, l1v, N);

  // Pairwise stage: 128 batches x 6 j-blocks
  k_pairwise<<<128 * 6, 256, 0, stream>>>(
      l0, l1v, positions,
      vv_W1, vv_b1, vv_W2, vv_b2,
      vr_W1, vr_b1, vr_W2, vr_b2,
      s_W1,  s_b1,  s_W2,  s_b2,
      h_W1,  h_b1,  h_W2,  h_b2,
      out);
}